// CDGS4_17592186044831
// MI455X (gfx1250) — compile-verified
//
#include <hip/hip_runtime.h>
#include <math.h>

// ---------------------------------------------------------------------------
// Sinkhorn (30 iters) + greedy unique argmax -> hard permutation, per 64x64.
// One wave32 per matrix. Sinkhorn kept in diag(r) * X0 * diag(c) form so the
// matrix is written once; row/col matvecs run on V_WMMA_F32_16X16X4_F32.
// Scale updates use v_rcp_f32 instead of IEEE division (argmax-safe).
// ---------------------------------------------------------------------------

typedef float v2f __attribute__((ext_vector_type(2)));
typedef float v4f __attribute__((ext_vector_type(4)));
typedef float v8f __attribute__((ext_vector_type(8)));

#define NDIM    64
#define PITCH   68                      // pad: keeps 16B row alignment, spreads banks
#define N_SINK  30
#define WPB     2                       // waves (= matrices) per block
#define MAT_F32 (NDIM * PITCH)          // 4352 floats
#define WAVE_F32 (MAT_F32 + 3 * NDIM)   // matrix + r + c + assign

static __device__ inline v8f wmma_f32_4(v2f a, v2f b, v8f c) {
  // D(16x16,f32) += A(16x4,f32) * B(4x16,f32)
  return __builtin_amdgcn_wmma_f32_16x16x4_f32(
      /*neg_a=*/false, a, /*neg_b=*/false, b,
      /*c_mod=*/(short)0, c, /*reuse_a=*/false, /*reuse_b=*/false);
}

static __device__ inline float fast_rcp(float x) {
  return __builtin_amdgcn_rcpf(x);      // v_rcp_f32, ~1 ulp
}

__global__ __launch_bounds__(WPB * 32)
void sinkhorn_perm_kernel(const float* __restrict__ logits,
                          const float* __restrict__ ptemp,
                          float* __restrict__ out, int nbat) {
  __shared__ float lds[WPB * WAVE_F32];

  const int lane  = threadIdx.x & 31;
  const int wave  = threadIdx.x >> 5;
  const int batch = blockIdx.x * WPB + wave;
  if (batch >= nbat) return;            // wave-uniform exit

  float* Mt  = lds + wave * WAVE_F32;   // 64 x PITCH matrix X0
  float* R   = Mt + MAT_F32;            // row scales r[64]
  float* C   = R + NDIM;                // col scales c[64]
  int*   ASG = (int*)(C + NDIM);        // assignment[64]

  const float inv_temp = 1.0f / (ptemp[0] + 1e-6f);
  const float* src = logits + (size_t)batch * (NDIM * NDIM);

  // ---- pass 1: coalesced load, scale, running max, stage into LDS --------
  float mx = -3.0e38f;
  #pragma unroll 8
  for (int i = 0; i < 32; ++i) {
    int e = (i * 32 + lane) * 4;        // linear element index
    v4f x = *(const v4f*)(src + e);
    x *= inv_temp;
    mx = fmaxf(mx, fmaxf(fmaxf(x.x, x.y), fmaxf(x.z, x.w)));
    int row = e >> 6, col = e & 63;
    *(v4f*)(Mt + row * PITCH + col) = x;
  }
  #pragma unroll
  for (int off = 16; off; off >>= 1) mx = fmaxf(mx, __shfl_xor(mx, off, 32));

  // ---- pass 2: X0 = exp(x - max); init r = c = 1 --------------------------
  #pragma unroll 4
  for (int i = 0; i < 32; ++i) {
    int e = (i * 32 + lane) * 4;
    int row = e >> 6, col = e & 63;
    v4f x = *(v4f*)(Mt + row * PITCH + col);
    x.x = __expf(x.x - mx); x.y = __expf(x.y - mx);
    x.z = __expf(x.z - mx); x.w = __expf(x.w - mx);
    *(v4f*)(Mt + row * PITCH + col) = x;
  }
  R[lane] = 1.0f; R[lane + 32] = 1.0f;
  C[lane] = 1.0f; C[lane + 32] = 1.0f;

  const int half = lane >> 4;           // 0: lanes 0-15, 1: lanes 16-31
  const int lm   = lane & 15;

  // ---- Sinkhorn: r,c updates via WMMA matvecs -----------------------------
  for (int it = 0; it < N_SINK; ++it) {
    // row phase: t = X0 * c  (broadcast c across N). 4 row-blocks.
    #pragma unroll
    for (int ib = 0; ib < 4; ++ib) {
      v8f acc = {0.f, 0.f, 0.f, 0.f, 0.f, 0.f, 0.f, 0.f};
      const float* arow = Mt + (ib * 16 + lm) * PITCH + 2 * half;
      #pragma unroll
      for (int k0 = 0; k0 < 64; k0 += 4) {
        v2f a = *(const v2f*)(arow + k0);              // A layout: M=lm, K=2*half+v
        v2f b = *(const v2f*)(C + k0 + 2 * half);      // B[k,n] = c[k] (all n)
        acc = wmma_f32_4(a, b, acc);
      }
      // acc[v] = t[ib*16 + 8*half + v] (columns redundant); lanes 0/16 update r
      if (lm == 0) {
        #pragma unroll
        for (int v = 0; v < 8; ++v) {
          int m = ib * 16 + 8 * half + v;
          float rv = R[m];
          R[m] = rv * fast_rcp(fmaf(rv, acc[v], 1e-8f));
        }
      }
    }
    // col phase: u = r^T * X0 (broadcast r across M). 4 col-blocks.
    #pragma unroll
    for (int jb = 0; jb < 4; ++jb) {
      v8f acc = {0.f, 0.f, 0.f, 0.f, 0.f, 0.f, 0.f, 0.f};
      #pragma unroll
      for (int k0 = 0; k0 < 64; k0 += 4) {
        v2f a = *(const v2f*)(R + k0 + 2 * half);      // A[m,k] = r[k] (all m)
        int kk = k0 + 2 * half;
        v2f b;
        b.x = Mt[(kk + 0) * PITCH + jb * 16 + lm];     // B layout: K=2*half+v, N=lm
        b.y = Mt[(kk + 1) * PITCH + jb * 16 + lm];
        acc = wmma_f32_4(a, b, acc);
      }
      // acc[*] = u[jb*16 + lm] (rows redundant); lanes 0-15 update c
      if (half == 0) {
        int n = jb * 16 + lm;
        float cv = C[n];
        C[n] = cv * fast_rcp(fmaf(cv, acc[0], 1e-8f));
      }
    }
  }

  // ---- materialize s = diag(r) X0 diag(c) in place ------------------------
  #pragma unroll 4
  for (int i = 0; i < 32; ++i) {
    int e = (i * 32 + lane) * 4;
    int row = e >> 6, col = e & 63;
    float rr = R[row];
    v4f cc = *(v4f*)(C + col);
    v4f x  = *(v4f*)(Mt + row * PITCH + col);
    x *= rr;
    x *= cc;
    *(v4f*)(Mt + row * PITCH + col) = x;
  }

  // ---- greedy unique argmax: incremental per-row bests --------------------
  float rbv[2]; int rbc[2];             // each lane owns rows lane, lane+32
  #pragma unroll
  for (int h = 0; h < 2; ++h) {
    int row = lane + 32 * h;
    float best = -1.0f; int bc = 0;
    for (int c4 = 0; c4 < 64; c4 += 4) {
      v4f v = *(v4f*)(Mt + row * PITCH + c4);
      if (v.x > best) { best = v.x; bc = c4 + 0; }
      if (v.y > best) { best = v.y; bc = c4 + 1; }
      if (v.z > best) { best = v.z; bc = c4 + 2; }
      if (v.w > best) { best = v.w; bc = c4 + 3; }
    }
    rbv[h] = best; rbc[h] = bc;
  }
  unsigned long long used_r = 0ull, used_c = 0ull;
  for (int round = 0; round < 64; ++round) {
    float cv = -1.0f; int crow = 64, ccol = 64;
    #pragma unroll
    for (int h = 0; h < 2; ++h) {
      int row = lane + 32 * h;
      if (!((used_r >> row) & 1ull) && rbv[h] > cv) {
        cv = rbv[h]; crow = row; ccol = rbc[h];
      }
    }
    #pragma unroll
    for (int off = 16; off; off >>= 1) {       // wave argmax (deterministic ties)
      float ov = __shfl_xor(cv,   off, 32);
      int  orw = __shfl_xor(crow, off, 32);
      int  ocl = __shfl_xor(ccol, off, 32);
      if (ov > cv || (ov == cv && orw < crow)) { cv = ov; crow = orw; ccol = ocl; }
    }
    used_r |= 1ull << crow;
    used_c |= 1ull << ccol;
    if (lane == 0) ASG[crow] = ccol;
    // rescan rows whose best column was just taken
    #pragma unroll
    for (int h = 0; h < 2; ++h) {
      int row = lane + 32 * h;
      if (!((used_r >> row) & 1ull) && rbc[h] == ccol) {
        float best = -1.0f; int bc = 0;
        for (int c4 = 0; c4 < 64; c4 += 4) {
          v4f v = *(v4f*)(Mt + row * PITCH + c4);
          if (!((used_c >> (c4 + 0)) & 1ull) && v.x > best) { best = v.x; bc = c4 + 0; }
          if (!((used_c >> (c4 + 1)) & 1ull) && v.y > best) { best = v.y; bc = c4 + 1; }
          if (!((used_c >> (c4 + 2)) & 1ull) && v.z > best) { best = v.z; bc = c4 + 2; }
          if (!((used_c >> (c4 + 3)) & 1ull) && v.w > best) { best = v.w; bc = c4 + 3; }
        }
        rbv[h] = best; rbc[h] = bc;
      }
    }
  }

  // ---- emit hard permutation (straight-through forward), coalesced --------
  float* dst = out + (size_t)batch * (NDIM * NDIM);
  #pragma unroll 8
  for (int i = 0; i < 32; ++i) {
    int e = (i * 32 + lane) * 4;
    int row = e >> 6, col = e & 63;
    int a = ASG[row];
    v4f o;
    o.x = (a == col + 0) ? 1.0f : 0.0f;
    o.y = (a == col + 1) ? 1.0f : 0.0f;
    o.z = (a == col + 2) ? 1.0f : 0.0f;
    o.w = (a == col + 3) ? 1.0f : 0.0f;
    *(v4f*)(dst + e) = o;
  }
}

extern "C" void kernel_launch(void* const* d_in, const int* in_sizes, int n_in,
                              void* d_out, int out_size, void* d_ws, size_t ws_size,
                              hipStream_t stream) {
  const float* logits = (const float*)d_in[0];
  const float* ptemp  = (const float*)d_in[1];
  float* out = (float*)d_out;
  int nbat = in_sizes[0] / (NDIM * NDIM);   // 4096
  int blocks = (nbat + WPB - 1) / WPB;
  sinkhorn_perm_kernel<<<blocks, WPB * 32, 0, stream>>>(logits, ptemp, out, nbat);
}